// MultiBoxLoss_44049184587779
// MI455X (gfx1250) — compile-verified
//
#include <hip/hip_runtime.h>
#include <hip/hip_bf16.h>
#include <math.h>

typedef __attribute__((ext_vector_type(2))) float v2f;
typedef __attribute__((ext_vector_type(8))) float v8f;
typedef unsigned int       u32;
typedef unsigned long long u64;
typedef unsigned char      u8;

#define TPB   256
#define NWAVE (TPB / 32)
#define MAXOBJ 64
#define ELE   16   // elements per thread in streaming select kernels

// ---------------------------------------------------------------------------
// Wave-level sum via V_WMMA_F32_16X16X4_F32 (layout-agnostic):
//   A = ones(16x4)  -> D[m,n] = sum_k B[k,n]  (column sums of B, rows identical)
//   B = (v, 0) per lane -> each lane's v lands in exactly one (k,n) slot.
//   Horizontal add over the 16 columns of D (documented C/D layout) counts
//   every lane's value exactly once -> exact, order-fixed wave sum.
// Requires EXEC all ones (call sites are fully converged).
// ---------------------------------------------------------------------------
__device__ __forceinline__ float wave_sum_wmma(float v) {
  v2f a; a[0] = 1.0f; a[1] = 1.0f;
  v2f bm; bm[0] = v; bm[1] = 0.0f;
  v8f c = {};
  c = __builtin_amdgcn_wmma_f32_16x16x4_f32(false, a, false, bm, (short)0, c,
                                            false, false);
  float s = c[0];
  s += __shfl_xor(s, 1, 32);
  s += __shfl_xor(s, 2, 32);
  s += __shfl_xor(s, 4, 32);
  s += __shfl_xor(s, 8, 32);
  return s; // every lane holds the full 32-lane sum
}

// deterministic block reduction: n_pos via atomic (exact-integer float adds are
// order-independent), conf/loc partials written per block (reduced later in a
// fixed order)
__device__ __forceinline__ void block_red3(float v0, float v1, float v2,
                                           float* nposTarget, float* pConf,
                                           float* pLoc) {
  __shared__ float red[3][NWAVE];
  const int tid = threadIdx.x, wid = tid >> 5, lane = tid & 31;
  float s0 = wave_sum_wmma(v0);
  float s1 = wave_sum_wmma(v1);
  float s2 = wave_sum_wmma(v2);
  if (lane == 0) { red[0][wid] = s0; red[1][wid] = s1; red[2][wid] = s2; }
  __syncthreads();
  if (tid == 0) {
    float a0 = 0.f, a1 = 0.f, a2 = 0.f;
    for (int i = 0; i < NWAVE; ++i) { a0 += red[0][i]; a1 += red[1][i]; a2 += red[2][i]; }
    atomicAdd(nposTarget, a0);       // exact integers -> deterministic
    pConf[blockIdx.x] = a1;          // deterministic slot write
    pLoc[blockIdx.x]  = a2;
  }
  __syncthreads();
}

// ---------------------------------------------------------------------------
__global__ void k_zero(u32* __restrict__ p, unsigned n) {
  unsigned i = blockIdx.x * blockDim.x + threadIdx.x;
  if (i < n) p[i] = 0u;
}

// ---------------------------------------------------------------------------
// matching: per (b,p) compute IoU vs all objects; per-prior best (ov,obj);
// per-object best prior via packed u64 atomicMax (order-independent):
//   key = (bits(iou) << 32) | (0xFFFFFFFF - p)   (ties -> smallest p, as jnp)
// ---------------------------------------------------------------------------
__global__ void k_match(const float* __restrict__ priors,
                        const float* __restrict__ boxes,
                        float* __restrict__ ovBuf, u8* __restrict__ objBuf,
                        u64* __restrict__ keybuf, int P, int nObj) {
  const int cpb = (P + TPB - 1) / TPB;
  const int b = blockIdx.x / cpb;
  const int p = (blockIdx.x % cpb) * TPB + threadIdx.x;
  const bool valid = (p < P);

  __shared__ float sb[MAXOBJ * 6];
  __shared__ float sva[MAXOBJ];
  __shared__ u64 wk[NWAVE];
  if (threadIdx.x < nObj * 6)
    sb[threadIdx.x] = boxes[(size_t)b * nObj * 6 + threadIdx.x];
  __syncthreads();
  if (threadIdx.x < nObj) {
    const float* bb = &sb[threadIdx.x * 6];
    sva[threadIdx.x] = (bb[3] - bb[0]) * (bb[4] - bb[1]) * (bb[5] - bb[2]);
  }
  __syncthreads();

  float plx = 0, ply = 0, plz = 0, phx = 0, phy = 0, phz = 0, vb = 0;
  if (valid) {
    const float* pr = priors + (size_t)p * 6;
    plx = pr[0] - pr[3] * 0.5f; ply = pr[1] - pr[4] * 0.5f; plz = pr[2] - pr[5] * 0.5f;
    phx = pr[0] + pr[3] * 0.5f; phy = pr[1] + pr[4] * 0.5f; phz = pr[2] + pr[5] * 0.5f;
    vb = (phx - plx) * (phy - ply) * (phz - plz);
  }

  float bestOv = -1.0f; int bestObj = 0;
  const int lane = threadIdx.x & 31, wid = threadIdx.x >> 5;
  for (int o = 0; o < nObj; ++o) {
    u64 key = 0ull;
    if (valid) {
      const float* bb = &sb[o * 6];
      float ix = fmaxf(fminf(bb[3], phx) - fmaxf(bb[0], plx), 0.f);
      float iy = fmaxf(fminf(bb[4], phy) - fmaxf(bb[1], ply), 0.f);
      float iz = fmaxf(fminf(bb[5], phz) - fmaxf(bb[2], plz), 0.f);
      float inter = ix * iy * iz;
      float iou = inter / (sva[o] + vb - inter);
      if (iou > bestOv) { bestOv = iou; bestObj = o; }
      key = ((u64)__float_as_uint(iou) << 32) | (u64)(0xFFFFFFFFu - (u32)p);
    }
    for (int off = 16; off >= 1; off >>= 1) {
      u64 other = __shfl_xor(key, off, 32);
      if (other > key) key = other;
    }
    if (lane == 0) wk[wid] = key;
    __syncthreads();
    if (threadIdx.x == 0) {
      u64 m = wk[0];
      for (int i = 1; i < NWAVE; ++i) if (wk[i] > m) m = wk[i];
      atomicMax(&keybuf[b * nObj + o], m);
    }
    __syncthreads();
  }
  if (valid) {
    ovBuf[(size_t)b * P + p] = bestOv;
    objBuf[(size_t)b * P + p] = (u8)bestObj;
  }
}

// force each object's best prior to match that object (sequential per batch ->
// deterministic last-wins on duplicate priors, like the scatter in reference)
__global__ void k_force(const u64* __restrict__ keybuf, float* __restrict__ ovBuf,
                        u8* __restrict__ objBuf, int P, int nObj, int B) {
  int b = blockIdx.x * blockDim.x + threadIdx.x;
  if (b >= B) return;
  for (int o = 0; o < nObj; ++o) {
    u64 key = keybuf[b * nObj + o];
    u32 p = 0xFFFFFFFFu - (u32)(key & 0xFFFFFFFFull);
    ovBuf[(size_t)b * P + p] = 1.0f;
    objBuf[(size_t)b * P + p] = (u8)o;
  }
}

// ---------------------------------------------------------------------------
// per-prior CE + loc L1 partials; writes conf_neg array for the top-K select
// ---------------------------------------------------------------------------
__global__ void k_lossprep(const float* __restrict__ locs,
                           const float* __restrict__ scores,
                           const float* __restrict__ boxes,
                           const int* __restrict__ labels,
                           const float* __restrict__ priors,
                           const float* __restrict__ ovBuf,
                           const u8* __restrict__ objBuf,
                           float* __restrict__ cneg, float* __restrict__ nposF,
                           float* __restrict__ pConf, float* __restrict__ pLoc,
                           int P, int nObj, int C) {
  const int cpb = (P + TPB - 1) / TPB;
  const int b = blockIdx.x / cpb;
  const int p = (blockIdx.x % cpb) * TPB + threadIdx.x;
  const bool valid = (p < P);

  __shared__ float sb[MAXOBJ * 6];
  __shared__ int sl[MAXOBJ];
  if (threadIdx.x < nObj * 6)
    sb[threadIdx.x] = boxes[(size_t)b * nObj * 6 + threadIdx.x];
  if (threadIdx.x < nObj) sl[threadIdx.x] = labels[b * nObj + threadIdx.x];
  __syncthreads();

  float posf = 0.f, confPos = 0.f, locA = 0.f;
  if (valid) {
    const size_t idx = (size_t)b * P + p;
    float ov = ovBuf[idx];
    int o = (int)objBuf[idx];
    int cls = (ov < 0.5f) ? 0 : sl[o];
    const float* sc = scores + idx * C;
    float m = sc[0];
    for (int c = 1; c < C; ++c) m = fmaxf(m, sc[c]);
    float se = 0.f;
    for (int c = 0; c < C; ++c) se += __expf(sc[c] - m);
    float conf = (m + __logf(se)) - sc[cls];
    if (cls > 0) {
      posf = 1.f; confPos = conf;
      const float* bb = &sb[o * 6];
      const float* pr = priors + (size_t)p * 6;
      const float* pl = locs + idx * 6;
      float cx = (bb[0] + bb[3]) * 0.5f, cy = (bb[1] + bb[4]) * 0.5f,
            cz = (bb[2] + bb[5]) * 0.5f;
      float sx = bb[3] - bb[0], sy = bb[4] - bb[1], sz = bb[5] - bb[2];
      locA += fabsf(pl[0] - (cx - pr[0]) / (pr[3] / 10.0f));
      locA += fabsf(pl[1] - (cy - pr[1]) / (pr[4] / 10.0f));
      locA += fabsf(pl[2] - (cz - pr[2]) / (pr[5] / 10.0f));
      locA += fabsf(pl[3] - __logf(sx / pr[3]) * 5.0f);
      locA += fabsf(pl[4] - __logf(sy / pr[4]) * 5.0f);
      locA += fabsf(pl[5] - __logf(sz / pr[5]) * 5.0f);
      cneg[idx] = 0.f;
    } else {
      cneg[idx] = conf; // conf >= 0 always -> uint bit order == float order
    }
  }
  block_red3(posf, confPos, locA, &nposF[b], pConf, pLoc);
}

// deterministic final reduction of the per-block conf/loc partials
__global__ void k_sumred(const float* __restrict__ pConf,
                         const float* __restrict__ pLoc, int n,
                         float* __restrict__ cps, float* __restrict__ lsum) {
  float a = 0.f, b = 0.f;
  for (int i = threadIdx.x; i < n; i += TPB) { a += pConf[i]; b += pLoc[i]; }
  __shared__ float red[2][NWAVE];
  const int tid = threadIdx.x, wid = tid >> 5, lane = tid & 31;
  float wa = wave_sum_wmma(a);
  float wb = wave_sum_wmma(b);
  if (lane == 0) { red[0][wid] = wa; red[1][wid] = wb; }
  __syncthreads();
  if (tid == 0) {
    float ca = 0.f, cb = 0.f;
    for (int i = 0; i < NWAVE; ++i) { ca += red[0][i]; cb += red[1][i]; }
    cps[0] = ca; lsum[0] = cb;
  }
}

__global__ void k_initsel(const float* __restrict__ nposF, u32* __restrict__ prefix,
                          u32* __restrict__ kRem, int P, int B) {
  int b = blockIdx.x * blockDim.x + threadIdx.x;
  if (b >= B) return;
  int k = 3 * (int)(nposF[b] + 0.5f);
  if (k > P) k = P;
  prefix[b] = 0u;
  kRem[b] = (u32)k;
}

// radix-select histogram pass over conf_neg bit patterns (8 bits per pass)
__global__ void k_hist(const float* __restrict__ cneg,
                       const u32* __restrict__ prefix, u32* __restrict__ ghist,
                       int P, int shift) {
  const int cpb = (P + TPB * ELE - 1) / (TPB * ELE);
  const int b = blockIdx.x / cpb;
  const int chunk = blockIdx.x % cpb;
  __shared__ u32 h[256];
  h[threadIdx.x] = 0u;
  __syncthreads();
  const u32 pfx = prefix[b];
  const u32 maskHigh = (shift == 24) ? 0u : (0xFFFFFFFFu << (shift + 8));
  const int pbase = chunk * TPB * ELE + threadIdx.x;
  const size_t base = (size_t)b * P + pbase;
  for (int i = 0; i < ELE; ++i) {
    if (pbase + i * TPB < P) {
      u32 u = __float_as_uint(cneg[base + (size_t)i * TPB]);
      if ((u & maskHigh) == (pfx & maskHigh))
        atomicAdd(&h[(u >> shift) & 0xFFu], 1u);
    }
  }
  __syncthreads();
  u32 c = h[threadIdx.x];
  if (c) atomicAdd(&ghist[b * 256 + threadIdx.x], c); // integer -> deterministic
}

// per-batch scan: pick the bin containing the kRem-th largest, update prefix
__global__ void k_scan(u32* __restrict__ ghist, u32* __restrict__ prefix,
                       u32* __restrict__ kRem, int shift, int B) {
  int b = threadIdx.x;
  if (b >= B) return;
  u32 kr = kRem[b];
  u32 cum = 0; int sel = 0;
  for (int bin = 255; bin >= 0; --bin) {
    u32 c = ghist[b * 256 + bin];
    if (cum + c >= kr) { sel = bin; kr -= cum; break; }
    cum += c;
  }
  prefix[b] |= ((u32)sel) << shift;
  kRem[b] = kr;
  for (int i = 0; i < 256; ++i) ghist[b * 256 + i] = 0u; // ready for next pass
}

// per-block sum of strictly-greater-than-threshold values (ties in combine)
__global__ void k_topksum(const float* __restrict__ cneg,
                          const u32* __restrict__ prefix,
                          float* __restrict__ pHard, int P) {
  const int cpb = (P + TPB * ELE - 1) / (TPB * ELE);
  const int b = blockIdx.x / cpb;
  const int chunk = blockIdx.x % cpb;
  const u32 t = prefix[b];
  const int pbase = chunk * TPB * ELE + threadIdx.x;
  const size_t base = (size_t)b * P + pbase;
  float s = 0.f;
  for (int i = 0; i < ELE; ++i) {
    if (pbase + i * TPB < P) {
      float v = cneg[base + (size_t)i * TPB];
      if (__float_as_uint(v) > t) s += v;
    }
  }
  __shared__ float red[NWAVE];
  float w = wave_sum_wmma(s);
  const int lane = threadIdx.x & 31, wid = threadIdx.x >> 5;
  if (lane == 0) red[wid] = w;
  __syncthreads();
  if (threadIdx.x == 0) {
    float a = 0.f;
    for (int i = 0; i < NWAVE; ++i) a += red[i];
    pHard[blockIdx.x] = a; // deterministic slot write
  }
}

__global__ void k_combine(const float* __restrict__ nposF,
                          const float* __restrict__ pHard, int cpbE,
                          const u32* __restrict__ prefix,
                          const u32* __restrict__ kRem,
                          const float* __restrict__ confPosSum,
                          const float* __restrict__ locSum,
                          float* __restrict__ out, int B) {
  if (blockIdx.x == 0 && threadIdx.x == 0) {
    float tp = 0.f, hard = 0.f;
    for (int b = 0; b < B; ++b) {
      tp += nposF[b];
      float h = 0.f;
      for (int c = 0; c < cpbE; ++c) h += pHard[b * cpbE + c]; // fixed order
      u32 kr = kRem[b]; // #ties of threshold value included in top-K
      if (kr) h += (float)kr * __uint_as_float(prefix[b]);
      hard += h;
    }
    out[0] = (hard + confPosSum[0]) / tp;      // conf_loss
    out[1] = locSum[0] / (tp * 6.0f);          // loc_loss
  }
}

// ---------------------------------------------------------------------------
extern "C" void kernel_launch(void* const* d_in, const int* in_sizes, int n_in,
                              void* d_out, int out_size, void* d_ws,
                              size_t ws_size, hipStream_t stream) {
  const float* locs   = (const float*)d_in[0];
  const float* scores = (const float*)d_in[1];
  const float* boxes  = (const float*)d_in[2];
  const int*   labels = (const int*)d_in[3];
  const float* priors = (const float*)d_in[4];

  const int P    = in_sizes[4] / 6;
  const int B    = in_sizes[0] / (6 * P);
  const int nObj = in_sizes[3] / B;
  const int C    = in_sizes[1] / (B * P);

  const int cpb  = (P + TPB - 1) / TPB;
  const int cpbE = (P + TPB * ELE - 1) / (TPB * ELE);
  const int nBlk = B * cpb;

  // workspace layout
  char* ws = (char*)d_ws;
  size_t off = 0;
  auto take = [&](size_t bytes, size_t align) -> char* {
    off = (off + align - 1) & ~(align - 1);
    char* p = ws + off;
    off += bytes;
    return p;
  };
  float* ovBuf  = (float*)take((size_t)B * P * 4, 4);
  float* cneg   = (float*)take((size_t)B * P * 4, 4);
  u8*    objBuf = (u8*)take((size_t)B * P, 1);
  // zeroed small-state region [small0, smallEnd)
  char*  small0 = ws + ((off + 7) & ~(size_t)7);
  u64*   keybuf = (u64*)take((size_t)B * nObj * 8, 8);
  u32*   ghist  = (u32*)take((size_t)B * 256 * 4, 4);
  float* nposF  = (float*)take((size_t)B * 4, 4);
  u32*   prefix = (u32*)take((size_t)B * 4, 4);
  u32*   kRem   = (u32*)take((size_t)B * 4, 4);
  float* cps    = (float*)take(4, 4);
  float* lsum   = (float*)take(4, 4);
  char*  smallEnd = ws + off;
  // fully-overwritten partial arrays (no zeroing needed)
  float* pConf  = (float*)take((size_t)nBlk * 4, 4);
  float* pLoc   = (float*)take((size_t)nBlk * 4, 4);
  float* pHard  = (float*)take((size_t)B * cpbE * 4, 4);
  (void)ws_size;

  const unsigned smallWords = (unsigned)((smallEnd - small0) / 4);

  k_zero<<<(smallWords + TPB - 1) / TPB, TPB, 0, stream>>>((u32*)small0,
                                                           smallWords);
  k_match<<<B * cpb, TPB, 0, stream>>>(priors, boxes, ovBuf, objBuf, keybuf, P,
                                       nObj);
  k_force<<<1, TPB, 0, stream>>>(keybuf, ovBuf, objBuf, P, nObj, B);
  k_lossprep<<<B * cpb, TPB, 0, stream>>>(locs, scores, boxes, labels, priors,
                                          ovBuf, objBuf, cneg, nposF, pConf,
                                          pLoc, P, nObj, C);
  k_sumred<<<1, TPB, 0, stream>>>(pConf, pLoc, nBlk, cps, lsum);
  k_initsel<<<1, TPB, 0, stream>>>(nposF, prefix, kRem, P, B);
  for (int shift = 24; shift >= 0; shift -= 8) {
    k_hist<<<B * cpbE, TPB, 0, stream>>>(cneg, prefix, ghist, P, shift);
    k_scan<<<1, TPB, 0, stream>>>(ghist, prefix, kRem, shift, B);
  }
  k_topksum<<<B * cpbE, TPB, 0, stream>>>(cneg, prefix, pHard, P);
  k_combine<<<1, 32, 0, stream>>>(nposF, pHard, cpbE, prefix, kRem, cps, lsum,
                                  (float*)d_out, B);
}